// Morph2D_40175124087411
// MI455X (gfx1250) — compile-verified
//
#include <hip/hip_runtime.h>
#include <math.h>

#ifndef __has_builtin
#define __has_builtin(x) 0
#endif

#if __has_builtin(__builtin_amdgcn_global_load_async_to_lds_b32)
#define HAVE_ASYNC_LDS 1
#else
#define HAVE_ASYNC_LDS 0
#endif

static constexpr int IMW = 512;
static constexpr int IMH = 512;
static constexpr int NB  = 32;
static constexpr int TW  = 64;           // tile width  (output)
static constexpr int TH  = 16;           // tile height (output)
static constexpr int LW  = TW + 2;       // 66, LDS row stride (floats)
static constexpr int LH  = TH + 2;       // 18
static constexpr int NTHR = 256;         // 8 wave32s

#if HAVE_ASYNC_LDS
typedef __attribute__((address_space(1))) int* as1_iptr;
typedef __attribute__((address_space(3))) int* as3_iptr;
#endif

__device__ __forceinline__ void ce(float& a, float& b) {
  float lo = fminf(a, b);
  float hi = fmaxf(a, b);
  a = lo; b = hi;
}

// full 3-sorter in 3 VALU ops: v_min3_num_f32 / v_med3_f32 / v_max3_num_f32
__device__ __forceinline__ void sort3(float& a, float& b, float& c) {
  float lo  = fminf(fminf(a, b), c);
  float hi  = fmaxf(fmaxf(a, b), c);
  float mid = __builtin_amdgcn_fmed3f(a, b, c);
  a = lo; b = mid; c = hi;
}

__global__ __launch_bounds__(NTHR) void morph2d_soft_rank_kernel(
    const float* __restrict__ x,     // [B,H,W,1]
    const float* __restrict__ se,    // [9]
    const float* __restrict__ rank,  // [9]
    float* __restrict__ out) {       // [B,H,W,1]
  __shared__ float tile[LH * LW];
  __shared__ float wsm[9];

  const int tid = threadIdx.x;
  const int x0  = blockIdx.x * TW;
  const int y0  = blockIdx.y * TH;
  const int b   = blockIdx.z;
  const float* xb = x + (size_t)b * (IMH * IMW);

  // softmax(rank) once per block (cheap; avoids per-pixel transcendentals)
  if (tid == 0) {
    float m = rank[0];
#pragma unroll
    for (int i = 1; i < 9; ++i) m = fmaxf(m, rank[i]);
    float e[9];
    float s = 0.0f;
#pragma unroll
    for (int i = 0; i < 9; ++i) { e[i] = expf(rank[i] - m); s += e[i]; }
    float inv = 1.0f / s;
#pragma unroll
    for (int i = 0; i < 9; ++i) wsm[i] = e[i] * inv;
  }

  // ---- stage halo tile (66 x 18) into LDS ----
  const bool interior = (x0 > 0) && (x0 + TW < IMW) && (y0 > 0) && (y0 + TH < IMH);
  if (interior) {
#if HAVE_ASYNC_LDS
    for (int i = tid; i < LH * LW; i += NTHR) {
      const int r = i / LW;
      const int c = i - r * LW;
      const float* g = xb + (y0 - 1 + r) * IMW + (x0 - 1 + c);
      __builtin_amdgcn_global_load_async_to_lds_b32(
          (as1_iptr)g, (as3_iptr)(tile + i), 0, 0);
    }
#if __has_builtin(__builtin_amdgcn_s_wait_asynccnt)
    __builtin_amdgcn_s_wait_asynccnt(0);
#else
    asm volatile("s_wait_asynccnt 0" ::: "memory");
#endif
#else
    for (int i = tid; i < LH * LW; i += NTHR) {
      const int r = i / LW;
      const int c = i - r * LW;
      tile[i] = xb[(y0 - 1 + r) * IMW + (x0 - 1 + c)];
    }
#endif
  } else {
    for (int i = tid; i < LH * LW; i += NTHR) {
      const int r  = i / LW;
      const int c  = i - r * LW;
      const int gy = y0 - 1 + r;
      const int gx = x0 - 1 + c;
      float v = 0.0f;
      if (gy >= 0 && gy < IMH && gx >= 0 && gx < IMW) v = xb[gy * IMW + gx];
      tile[i] = v;
    }
  }
  __syncthreads();

  // uniform weights into registers
  const float s0 = se[0], s1 = se[1], s2 = se[2], s3 = se[3], s4 = se[4],
              s5 = se[5], s6 = se[6], s7 = se[7], s8 = se[8];
  const float w0 = wsm[0], w1 = wsm[1], w2 = wsm[2], w3 = wsm[3], w4 = wsm[4],
              w5 = wsm[5], w6 = wsm[6], w7 = wsm[7], w8 = wsm[8];

  const int tx = tid & (TW - 1);  // 0..63
  const int ty = tid >> 6;        // 0..3
  const int r0 = ty * 4;          // 4 CONSECUTIVE output rows r0..r0+3

  // Load the 6 shared window rows once: tile rows r0..r0+5, cols tx..tx+2.
  // 3 consecutive floats per row -> ds_load_b96; 4.5 LDS reads/pixel instead of 9.
  float c6[6][3];
  {
    const float* p = tile + r0 * LW + tx;
#pragma unroll
    for (int r = 0; r < 6; ++r) {
      c6[r][0] = p[r * LW + 0];
      c6[r][1] = p[r * LW + 1];
      c6[r][2] = p[r * LW + 2];
    }
  }

#pragma unroll
  for (int k = 0; k < 4; ++k) {
    float a0 = c6[k + 0][0] * s0;
    float a1 = c6[k + 0][1] * s1;
    float a2 = c6[k + 0][2] * s2;
    float a3 = c6[k + 1][0] * s3;
    float a4 = c6[k + 1][1] * s4;
    float a5 = c6[k + 1][2] * s5;
    float a6 = c6[k + 2][0] * s6;
    float a7 = c6[k + 2][1] * s7;
    float a8 = c6[k + 2][2] * s8;

    // 9-element sorting network: sort rows, sort columns, then fix-up tail.
    sort3(a0, a1, a2); sort3(a3, a4, a5); sort3(a6, a7, a8);   // rows
    sort3(a0, a3, a6); sort3(a1, a4, a7); sort3(a2, a5, a8);   // columns
    ce(a5, a7);
    ce(a1, a3);
    sort3(a2, a4, a6);   // replaces CE(2,6),CE(4,6),CE(2,4)
    ce(a5, a6);
    ce(a2, a3);

    float r = a0 * w0;
    r = fmaf(a1, w1, r);
    r = fmaf(a2, w2, r);
    r = fmaf(a3, w3, r);
    r = fmaf(a4, w4, r);
    r = fmaf(a5, w5, r);
    r = fmaf(a6, w6, r);
    r = fmaf(a7, w7, r);
    r = fmaf(a8, w8, r);

    out[((size_t)b * IMH + (y0 + r0 + k)) * IMW + (x0 + tx)] = r;
  }
}

extern "C" void kernel_launch(void* const* d_in, const int* in_sizes, int n_in,
                              void* d_out, int out_size, void* d_ws, size_t ws_size,
                              hipStream_t stream) {
  (void)in_sizes; (void)n_in; (void)out_size; (void)d_ws; (void)ws_size;
  const float* x    = (const float*)d_in[0];
  const float* se   = (const float*)d_in[1];
  const float* rank = (const float*)d_in[2];
  float* out        = (float*)d_out;

  dim3 grid(IMW / TW, IMH / TH, NB);   // 8 x 32 x 32 = 8192 blocks
  morph2d_soft_rank_kernel<<<grid, NTHR, 0, stream>>>(x, se, rank, out);
}